// WeakREChead_26585847562341
// MI455X (gfx1250) — compile-verified
//
#include <hip/hip_runtime.h>

typedef __attribute__((ext_vector_type(2))) float v2f;
typedef __attribute__((ext_vector_type(8))) float v8f;

#define B_DIM 256
#define V_DIM 4096
#define D_DIM 64
#define NCHUNK 8
#define VCHUNK (V_DIM / NCHUNK)   // 512 v per block

// ---------------------------------------------------------------------------
// Kernel 1: fused f32 WMMA GEMM (M=256 lan rows x K=64 x N=v) + per-(b,a)
// running top-2 over v. wave32 layout per CDNA5 ISA 7.12.2.
// grid = (a, chunk); block = 256 threads = 8 waves; wave w -> b rows [32w,32w+32)
// Double-buffered B fragments so global_load_b64s overlap the WMMA stream.
// ---------------------------------------------------------------------------
__global__ __launch_bounds__(256) void wrec_gemm_top2(
    const float* __restrict__ fusion,   // [256, 4096, 64]
    const float* __restrict__ lan,      // [256, 1, 64]
    float* __restrict__ partial)        // [B][B][NCHUNK][2]
{
    const int a     = blockIdx.x;
    const int chunk = blockIdx.y;
    const int tid   = threadIdx.x;
    const int wave  = tid >> 5;
    const int lane  = tid & 31;
    const int n     = lane & 15;        // N (and A-matrix M) index within tile
    const int hi    = lane >> 4;        // wave half
    const int koff  = hi * 2;           // K striping: half0 -> K=0,1; half1 -> K=2,3

    const int b0 = wave * 32;

    // Preload A fragments (lan) for both 16-row tiles: 16 K-steps x float2 each.
    v2f a0[16], a1[16];
    {
        const float* lrow0 = lan + (b0 + n) * D_DIM + koff;
        const float* lrow1 = lan + (b0 + 16 + n) * D_DIM + koff;
#pragma unroll
        for (int k = 0; k < 16; ++k) {
            a0[k] = *(const v2f*)(lrow0 + k * 4);
            a1[k] = *(const v2f*)(lrow1 + k * 4);
        }
    }

    // Per-lane running top-2 for each of the 8 C rows, both tiles.
    float t1a[8], t2a[8], t1b[8], t2b[8];
#pragma unroll
    for (int r = 0; r < 8; ++r) {
        t1a[r] = -__builtin_inff(); t2a[r] = -__builtin_inff();
        t1b[r] = -__builtin_inff(); t2b[r] = -__builtin_inff();
    }

    const int    v0base = chunk * VCHUNK;
    const size_t aBase  = (size_t)a * V_DIM;

    // ---- double-buffered subtile pipeline ----------------------------------
    v2f bufA[16], bufB[16];

    auto loadBuf = [&](v2f (&buf)[16], int vs) {
        const float* frow = fusion + (aBase + v0base + vs + n) * D_DIM + koff;
        __builtin_prefetch(frow + 32 * D_DIM, 0, 1);   // 2 subtiles ahead
#pragma unroll
        for (int k = 0; k < 16; ++k) buf[k] = *(const v2f*)(frow + k * 4);
    };

    auto computeBuf = [&](const v2f (&buf)[16]) {
        v8f c0 = {0.f,0.f,0.f,0.f,0.f,0.f,0.f,0.f};
        v8f c1 = {0.f,0.f,0.f,0.f,0.f,0.f,0.f,0.f};
#pragma unroll
        for (int k = 0; k < 16; ++k) {
            c0 = __builtin_amdgcn_wmma_f32_16x16x4_f32(
                     false, a0[k], false, buf[k], (short)0, c0, false, false);
            c1 = __builtin_amdgcn_wmma_f32_16x16x4_f32(
                     false, a1[k], false, buf[k], (short)0, c1, false, false);
        }
        // branchless running top-2 (16 sim values per lane)
#pragma unroll
        for (int r = 0; r < 8; ++r) {
            float v = c0[r];
            t2a[r] = fmaxf(t2a[r], fminf(t1a[r], v));
            t1a[r] = fmaxf(t1a[r], v);
            v = c1[r];
            t2b[r] = fmaxf(t2b[r], fminf(t1b[r], v));
            t1b[r] = fmaxf(t1b[r], v);
        }
    };

    loadBuf(bufA, 0);
#pragma unroll 1
    for (int vs = 0; vs < VCHUNK; vs += 32) {
        loadBuf(bufB, vs + 16);            // overlap with compute on bufA
        computeBuf(bufA);
        if (vs + 32 < VCHUNK)              // uniform guard: skip OOB tail load
            loadBuf(bufA, vs + 32);
        computeBuf(bufB);
    }

    // Merge (top1,top2) across the 16 lanes of each half (N columns).
    // xor masks 1,2,4,8 never cross the half boundary (bit 4 untouched).
#pragma unroll
    for (int r = 0; r < 8; ++r) {
#pragma unroll
        for (int m = 1; m <= 8; m <<= 1) {
            float o1 = __shfl_xor(t1a[r], m, 32);
            float o2 = __shfl_xor(t2a[r], m, 32);
            float n1 = fmaxf(t1a[r], o1);
            float n2 = fmaxf(fminf(t1a[r], o1), fmaxf(t2a[r], o2));
            t1a[r] = n1; t2a[r] = n2;
            o1 = __shfl_xor(t1b[r], m, 32);
            o2 = __shfl_xor(t2b[r], m, 32);
            n1 = fmaxf(t1b[r], o1);
            n2 = fmaxf(fminf(t1b[r], o1), fmaxf(t2b[r], o2));
            t1b[r] = n1; t2b[r] = n2;
        }
    }

    // Lane 0 of each half writes 8 rows per tile.
    if (n == 0) {
#pragma unroll
        for (int r = 0; r < 8; ++r) {
            int    ba  = b0 + r + hi * 8;                 // tile 0 rows
            size_t idx = (((size_t)ba * B_DIM + a) * NCHUNK + chunk) * 2;
            partial[idx]     = t1a[r];
            partial[idx + 1] = t2a[r];
            int    bb  = b0 + 16 + r + hi * 8;            // tile 1 rows
            idx = (((size_t)bb * B_DIM + a) * NCHUNK + chunk) * 2;
            partial[idx]     = t1b[r];
            partial[idx + 1] = t2b[r];
        }
    }
}

// ---------------------------------------------------------------------------
// Kernel 2: fold NCHUNK disjoint (max, second-max) pairs into a global pair.
// ---------------------------------------------------------------------------
__global__ __launch_bounds__(256) void wrec_merge_chunks(
    const float* __restrict__ partial,
    float* __restrict__ T1, float* __restrict__ T2)
{
    int idx = blockIdx.x * blockDim.x + threadIdx.x;   // b*256 + a
    if (idx >= B_DIM * B_DIM) return;
    const float* p = partial + (size_t)idx * NCHUNK * 2;
    float g1 = -__builtin_inff(), g2 = -__builtin_inff();
#pragma unroll
    for (int c = 0; c < NCHUNK; ++c) {
        float c1 = p[c * 2 + 0];
        float c2 = p[c * 2 + 1];
        if (c1 > g1) { g2 = fmaxf(g1, c2); g1 = c1; }
        else         { g2 = fmaxf(g2, c1); }
    }
    T1[idx] = g1;
    T2[idx] = g2;
}

// ---------------------------------------------------------------------------
// Kernel 3: per-row 511-logit log-softmax + mean CE. One block, thread b = row.
// logits[b] = [ T1[b,0..255], T2[b, j+(j>=b)] for j in 0..254 ];
// label logit = T1[b,b] (diagonal of max_sim_0).
// ---------------------------------------------------------------------------
__global__ __launch_bounds__(256) void wrec_loss(
    const float* __restrict__ T1, const float* __restrict__ T2,
    float* __restrict__ out)
{
    const int b = threadIdx.x;
    const float* r1 = T1 + b * B_DIM;
    const float* r2 = T2 + b * B_DIM;

    float mx = -__builtin_inff();
    for (int j = 0; j < B_DIM; ++j) mx = fmaxf(mx, r1[j]);
    for (int j = 0; j < B_DIM - 1; ++j) {
        int col = j + (j >= b);
        mx = fmaxf(mx, r2[col]);
    }
    float s = 0.f;
    for (int j = 0; j < B_DIM; ++j) s += expf(r1[j] - mx);
    for (int j = 0; j < B_DIM - 1; ++j) {
        int col = j + (j >= b);
        s += expf(r2[col] - mx);
    }
    float loss = -(r1[b] - (mx + logf(s)));

    __shared__ float red[B_DIM];
    red[b] = loss;
    __syncthreads();
    for (int off = 128; off > 0; off >>= 1) {
        if (b < off) red[b] += red[b + off];
        __syncthreads();
    }
    if (b == 0) out[0] = red[0] / (float)B_DIM;
}

// ---------------------------------------------------------------------------
extern "C" void kernel_launch(void* const* d_in, const int* in_sizes, int n_in,
                              void* d_out, int out_size, void* d_ws, size_t ws_size,
                              hipStream_t stream)
{
    (void)in_sizes; (void)n_in; (void)out_size; (void)ws_size;
    const float* fusion = (const float*)d_in[0];   // [256,4096,64] f32
    const float* lan    = (const float*)d_in[1];   // [256,1,64]    f32
    float*       out    = (float*)d_out;           // scalar f32

    float* partial = (float*)d_ws;                                   // B*B*NCHUNK*2 f32 (4 MB)
    float* T1      = partial + (size_t)B_DIM * B_DIM * NCHUNK * 2;   // B*B f32
    float* T2      = T1 + (size_t)B_DIM * B_DIM;                     // B*B f32

    dim3 grid1(B_DIM, NCHUNK);
    wrec_gemm_top2<<<grid1, 256, 0, stream>>>(fusion, lan, partial);
    wrec_merge_chunks<<<(B_DIM * B_DIM) / 256, 256, 0, stream>>>(partial, T1, T2);
    wrec_loss<<<1, 256, 0, stream>>>(T1, T2, out);
}